// FuzzyPooling_55155970015959
// MI455X (gfx1250) — compile-verified
//
#include <hip/hip_runtime.h>
#include <stdint.h>

// FuzzyPooling (2x2, stride 2) on (32,64,128,128) f32 -> (32,64,64,64) f32.
// Bandwidth-bound: stream tiles HBM -> LDS with CDNA5 async-to-LDS engine
// (global_load_async_to_lds_b128 / s_wait_asynccnt), double-buffered.

#define TILE_IN_BYTES  8192u      // 16 input rows * 128 floats * 4B (contiguous)
#define TILE_IN_FLOATS 2048
#define TILE_OUT_FLOATS 512       // 8 output rows * 64 floats
#define THREADS 256

__device__ __forceinline__ void issue_tile_async(const float* __restrict__ x,
                                                 int tile, unsigned lds_off,
                                                 unsigned off0, unsigned off1) {
    const char* g = (const char*)x + (size_t)tile * TILE_IN_BYTES;
    // VDST = 32-bit LDS byte address, VADDR = 64-bit global address.
    asm volatile("global_load_async_to_lds_b128 %0, %1, off"
                 :: "v"(lds_off + off0), "v"(g + off0) : "memory");
    asm volatile("global_load_async_to_lds_b128 %0, %1, off"
                 :: "v"(lds_off + off1), "v"(g + off1) : "memory");
}

// One 2x2 patch of the fuzzy pool. Matches the reference semantics:
// mu1: tri(v,1.5,1.5)  mu2: tri(v,3.0,1.5)  mu3: tri(v,3.0,1.5)
// sel = argmax(sum mu_i) with ties -> lowest index; out = den==0 ? 0 : num/den.
__device__ __forceinline__ float fuzzy_patch(float v0, float v1, float v2, float v3) {
    const float iw = 1.0f / 1.5f;
    float a10 = fmaxf(1.0f - fabsf(v0 - 1.5f) * iw, 0.0f);
    float a11 = fmaxf(1.0f - fabsf(v1 - 1.5f) * iw, 0.0f);
    float a12 = fmaxf(1.0f - fabsf(v2 - 1.5f) * iw, 0.0f);
    float a13 = fmaxf(1.0f - fabsf(v3 - 1.5f) * iw, 0.0f);
    float a20 = fmaxf(1.0f - fabsf(v0 - 3.0f) * iw, 0.0f);
    float a21 = fmaxf(1.0f - fabsf(v1 - 3.0f) * iw, 0.0f);
    float a22 = fmaxf(1.0f - fabsf(v2 - 3.0f) * iw, 0.0f);
    float a23 = fmaxf(1.0f - fabsf(v3 - 3.0f) * iw, 0.0f);
    // mu3 numerically identical to mu2 (centers/widths coincide in the reference)
    float s1 = ((a10 + a11) + (a12 + a13));
    float s2 = ((a20 + a21) + (a22 + a23));
    float s3 = s2;
    bool p1 = (s1 >= s2) && (s1 >= s3);   // argmax tie -> index 0 first
    bool p2 = (s2 >= s3);                 // then index 1 over 2
    float u0 = p1 ? a10 : (p2 ? a20 : a20);
    float u1 = p1 ? a11 : (p2 ? a21 : a21);
    float u2 = p1 ? a12 : (p2 ? a22 : a22);
    float u3 = p1 ? a13 : (p2 ? a23 : a23);
    float num = ((u0 * v0 * v0 + u1 * v1 * v1) + (u2 * v2 * v2 + u3 * v3 * v3));
    float den = ((u0 * v0 + u1 * v1) + (u2 * v2 + u3 * v3));
    return (den == 0.0f) ? 0.0f : (num / den);
}

__global__ __launch_bounds__(THREADS)
void FuzzyPooling_55155970015959_kernel(const float* __restrict__ x,
                                        float* __restrict__ out,
                                        int n_tiles) {
    __shared__ __align__(16) float lds[2 * TILE_IN_FLOATS];   // 16 KB double buffer

    const int tid = threadIdx.x;
    const unsigned lds_base = (unsigned)(uintptr_t)(&lds[0]); // low 32 bits = LDS offset
    // Two b128 async-copy slots per thread per tile: 512 * 16B = 8 KB tile
    const unsigned off0 = (unsigned)tid * 16u;
    const unsigned off1 = off0 + 256u * 16u;

    const int stride = (int)gridDim.x;
    int t = (int)blockIdx.x;
    if (t < n_tiles) issue_tile_async(x, t, lds_base, off0, off1);

    // Compute-side coordinates within a tile (8 out rows x 64 cols, 2 cols/thread)
    const int r  = tid >> 5;      // output row 0..7
    const int c2 = tid & 31;      // output column pair 0..31
    const float* lrow0base = &lds[(2 * r) * 128 + 4 * c2];
    const float* lrow1base = &lds[(2 * r + 1) * 128 + 4 * c2];

    int buf = 0;
    for (; t < n_tiles; t += stride) {
        const int nt = t + stride;
        if (nt < n_tiles) {
            issue_tile_async(x, nt, lds_base + (unsigned)(buf ^ 1) * TILE_IN_BYTES,
                             off0, off1);
            asm volatile("s_wait_asynccnt 2" ::: "memory");  // tile t landed (this wave)
        } else {
            asm volatile("s_wait_asynccnt 0" ::: "memory");
        }
        __syncthreads();   // all waves' async copies of tile t visible in LDS

        const float4 a = *(const float4*)(lrow0base + buf * TILE_IN_FLOATS); // ds_load_b128
        const float4 b = *(const float4*)(lrow1base + buf * TILE_IN_FLOATS); // ds_load_b128

        float2 o;
        o.x = fuzzy_patch(a.x, a.y, b.x, b.y);
        o.y = fuzzy_patch(a.z, a.w, b.z, b.w);
        *(float2*)(out + (size_t)t * TILE_OUT_FLOATS + r * 64 + 2 * c2) = o;

        __syncthreads();   // buffer 'buf' free before it is refilled next iteration
        buf ^= 1;
    }
}

extern "C" void kernel_launch(void* const* d_in, const int* in_sizes, int n_in,
                              void* d_out, int out_size, void* d_ws, size_t ws_size,
                              hipStream_t stream) {
    (void)n_in; (void)out_size; (void)d_ws; (void)ws_size;
    const float* x = (const float*)d_in[0];
    float* out = (float*)d_out;
    const int n = in_sizes[0];                 // 32*64*128*128
    const int n_tiles = n / TILE_IN_FLOATS;    // 16384
    int grid = n_tiles < 2048 ? n_tiles : 2048;
    if (grid < 1) grid = 1;
    FuzzyPooling_55155970015959_kernel<<<grid, THREADS, 0, stream>>>(x, out, n_tiles);
}